// TreeLSTM_43611097923955
// MI455X (gfx1250) — compile-verified
//
#include <hip/hip_runtime.h>
#include <hip/hip_bf16.h>

typedef __attribute__((ext_vector_type(2))) float v2f;
typedef __attribute__((ext_vector_type(8))) float v8f;

#define BATCH  256
#define LEAVES 256
#define NNODES 511
#define HID    256
#define DINP   128
#define NCOLS  1280   // packed output columns: [iou(768) | f0(256) | f1(256)]
#define KROWS  640    // packed K rows: [x(128) | h1(256) | h2(256)]
#define LDSTR  1284   // LDS row stride (floats); 8*1284 % 64 == 32 -> no bank clash

// ---------------------------------------------------------------------------
// Pack weights into a single K=640 x N=1280 matrix in WMMA-B-operand layout:
// Wt[(k>>2)*NCOLS*4 + n*4 + (k&3)] -> each lane's B fetch is a contiguous
// float2 (K = k0+2*half + {0,1}) at its column. Folds all biases. Also
// zero-fills a 128-float "zero embedding row" used for input_ids == 0.
// ---------------------------------------------------------------------------
__global__ void pack_kernel(
    const float* __restrict__ Wioux,  const float* __restrict__ bioux,
    const float* __restrict__ Wiouh1, const float* __restrict__ biouh1,
    const float* __restrict__ Wiouh2, const float* __restrict__ biouh2,
    const float* __restrict__ Wfx,    const float* __restrict__ bfx,
    const float* __restrict__ Wfh11,  const float* __restrict__ bfh11,
    const float* __restrict__ Wfh12,  const float* __restrict__ bfh12,
    const float* __restrict__ Wfh21,  const float* __restrict__ bfh21,
    const float* __restrict__ Wfh22,  const float* __restrict__ bfh22,
    float* __restrict__ Wt, float* __restrict__ biasc, float* __restrict__ zerox)
{
    int idx = blockIdx.x * blockDim.x + threadIdx.x;
    if (idx < KROWS * NCOLS) {
        int k = idx / NCOLS, n = idx % NCOLS;
        float v;
        if (k < DINP) {
            if (n < 768)        v = Wioux[k * 768 + n];
            else if (n < 1024)  v = Wfx[k * 256 + (n - 768)];
            else                v = Wfx[k * 256 + (n - 1024)];
        } else if (k < DINP + HID) {
            int kk = k - DINP;
            if (n < 768)        v = Wiouh1[kk * 768 + n];
            else if (n < 1024)  v = Wfh11[kk * 256 + (n - 768)];
            else                v = Wfh21[kk * 256 + (n - 1024)];
        } else {
            int kk = k - DINP - HID;
            if (n < 768)        v = Wiouh2[kk * 768 + n];
            else if (n < 1024)  v = Wfh12[kk * 256 + (n - 768)];
            else                v = Wfh22[kk * 256 + (n - 1024)];
        }
        Wt[(size_t)(k >> 2) * (NCOLS * 4) + n * 4 + (k & 3)] = v;
    }
    if (idx < NCOLS) {
        int n = idx;
        float bv;
        if (n < 768)        bv = bioux[n] + biouh1[n] + biouh2[n];
        else if (n < 1024)  bv = bfx[n - 768]  + bfh11[n - 768]  + bfh12[n - 768];
        else                bv = bfx[n - 1024] + bfh21[n - 1024] + bfh22[n - 1024];
        biasc[n] = bv;
    }
    if (idx < DINP) zerox[idx] = 0.0f;
}

__device__ __forceinline__ float sigmoidf_(float x) {
    return 1.0f / (1.0f + __expf(-x));
}

// ---------------------------------------------------------------------------
// Fused level kernel: block = 8 waves, owns 16 output rows x ALL columns.
// Wave w computes TPW 16x16 tiles (cols [w*TPW*16, (w+1)*TPW*16)) with
// V_WMMA_F32_16X16X4_F32; preact staged in LDS; gate math + h/c update done
// in-block (no preact round-trip to HBM).
//   A row m: cols [0,128)   = emb[input_ids[b, node_off+j]] (id 0 -> zero row)
//            cols [128,384) = h_prev row 2*m   (children are contiguous pair)
//            cols [384,640) = h_prev row 2*m+1
// ---------------------------------------------------------------------------
template <int TPW, bool INTERNAL>
__global__ __launch_bounds__(256) void fused_level_kernel(
    const int*   __restrict__ input_ids,
    const float* __restrict__ emb,
    const float* __restrict__ zerox,
    const float* __restrict__ h_prev,
    const float* __restrict__ c_prev,
    const float* __restrict__ Wt,
    const float* __restrict__ biasc,
    float*       __restrict__ h_out,
    float*       __restrict__ c_out,
    int node_off, int log2n, int n_nodes)
{
    __shared__ float pre[16 * LDSTR];   // 82 KB of the 320 KB WGP LDS

    const int wave  = threadIdx.x >> 5;
    const int lane  = threadIdx.x & 31;
    const int half2 = (lane >> 4) << 1;   // 0 or 2: K sub-offset for this half
    const int l15   = lane & 15;
    const int m0    = blockIdx.x * 16;
    const int mrow  = m0 + l15;

    // x-gather source (emb row forced to zero row when id == 0)
    const int bidx = mrow >> log2n;
    const int j    = mrow & (n_nodes - 1);
    const int id   = input_ids[bidx * NNODES + node_off + j];
    const float* px  = (id == 0) ? zerox : (emb + (size_t)id * DINP);
    const float* ph1 = h_prev + (size_t)mrow * (2 * HID);   // row 2*mrow
    const float* ph2 = ph1 + HID;                           // row 2*mrow+1

    const int colb = wave * (TPW * 16) + l15;   // this wave's first col + lane

    v8f acc[TPW] = {};

    // ---- x segment (k = 0..127), no branches, single b64 A load/iter -----
    for (int k0 = 0; k0 < DINP; k0 += 4) {
        v2f a = *(const v2f*)(px + k0 + half2);
        const float* wb = Wt + (size_t)(k0 >> 2) * (NCOLS * 4) + half2;
        #pragma unroll
        for (int t = 0; t < TPW; ++t) {
            v2f bv = *(const v2f*)(wb + (colb + t * 16) * 4);
            acc[t] = __builtin_amdgcn_wmma_f32_16x16x4_f32(
                false, a, false, bv, (short)0, acc[t], false, false);
        }
    }
    if (INTERNAL) {
        // ---- h1 segment (k = 128..383) -----------------------------------
        for (int k0 = DINP; k0 < DINP + HID; k0 += 4) {
            v2f a = *(const v2f*)(ph1 + (k0 - DINP + half2));
            const float* wb = Wt + (size_t)(k0 >> 2) * (NCOLS * 4) + half2;
            #pragma unroll
            for (int t = 0; t < TPW; ++t) {
                v2f bv = *(const v2f*)(wb + (colb + t * 16) * 4);
                acc[t] = __builtin_amdgcn_wmma_f32_16x16x4_f32(
                    false, a, false, bv, (short)0, acc[t], false, false);
            }
        }
        // ---- h2 segment (k = 384..639) -----------------------------------
        for (int k0 = DINP + HID; k0 < KROWS; k0 += 4) {
            v2f a = *(const v2f*)(ph2 + (k0 - DINP - HID + half2));
            const float* wb = Wt + (size_t)(k0 >> 2) * (NCOLS * 4) + half2;
            #pragma unroll
            for (int t = 0; t < TPW; ++t) {
                v2f bv = *(const v2f*)(wb + (colb + t * 16) * 4);
                acc[t] = __builtin_amdgcn_wmma_f32_16x16x4_f32(
                    false, a, false, bv, (short)0, acc[t], false, false);
            }
        }
    }

    // ---- stage preact tile in LDS (C layout: row = v + 8*half) -----------
    #pragma unroll
    for (int t = 0; t < TPW; ++t) {
        const int col = colb + t * 16;
        const float bv = biasc[col];
        #pragma unroll
        for (int v = 0; v < 8; ++v) {
            const int r = v + (half2 << 2);   // v + 8*half
            pre[r * LDSTR + col] = acc[t][v] + bv;
        }
    }
    __syncthreads();

    // ---- gate math for 16 rows x 256 hidden units ------------------------
    for (int e = threadIdx.x; e < 16 * HID; e += 256) {
        const int r = e >> 8;
        const int c = e & (HID - 1);
        const float* pr = pre + r * LDSTR;
        const float gi = sigmoidf_(pr[c]);
        const float go = sigmoidf_(pr[HID + c]);
        const float gu = tanhf(pr[2 * HID + c]);
        float cv = gi * gu;
        const int m = m0 + r;
        if (INTERNAL) {
            const float f0 = sigmoidf_(pr[3 * HID + c]);
            const float f1 = sigmoidf_(pr[4 * HID + c]);
            cv += f0 * c_prev[(size_t)(2 * m) * HID + c]
                + f1 * c_prev[(size_t)(2 * m + 1) * HID + c];
        }
        c_out[(size_t)m * HID + c] = cv;
        h_out[(size_t)m * HID + c] = go * tanhf(cv);
    }
}

// ---------------------------------------------------------------------------
// Tiny head: pred[b] = relu(root_h[b] @ Wl1 + bl1) @ Wl2 + bl2
// ---------------------------------------------------------------------------
__global__ void head_kernel(const float* __restrict__ root_h,
                            const float* __restrict__ Wl1,
                            const float* __restrict__ bl1,
                            const float* __restrict__ Wl2,
                            const float* __restrict__ bl2,
                            float* __restrict__ out)
{
    __shared__ float red[128];
    int b = blockIdx.x, t = threadIdx.x;
    float val = 0.0f;
    if (t < 100) {
        float s = bl1[t];
        const float* hr = root_h + (size_t)b * HID;
        for (int k = 0; k < HID; ++k) s += hr[k] * Wl1[k * 100 + t];
        s = fmaxf(s, 0.0f);
        val = s * Wl2[t];
    }
    red[t] = val;
    __syncthreads();
    for (int offc = 64; offc > 0; offc >>= 1) {
        if (t < offc) red[t] += red[t + offc];
        __syncthreads();
    }
    if (t == 0) out[b] = red[0] + bl2[0];
}

// ---------------------------------------------------------------------------
extern "C" void kernel_launch(void* const* d_in, const int* in_sizes, int n_in,
                              void* d_out, int out_size, void* d_ws, size_t ws_size,
                              hipStream_t stream)
{
    const int*   input_ids = (const int*)  d_in[0];
    const float* emb    = (const float*)d_in[1];
    const float* Wioux  = (const float*)d_in[2];
    const float* bioux  = (const float*)d_in[3];
    const float* Wiouh1 = (const float*)d_in[4];
    const float* biouh1 = (const float*)d_in[5];
    const float* Wiouh2 = (const float*)d_in[6];
    const float* biouh2 = (const float*)d_in[7];
    const float* Wfx    = (const float*)d_in[8];
    const float* bfx    = (const float*)d_in[9];
    const float* Wfh11  = (const float*)d_in[10];
    const float* bfh11  = (const float*)d_in[11];
    const float* Wfh12  = (const float*)d_in[12];
    const float* bfh12  = (const float*)d_in[13];
    const float* Wfh21  = (const float*)d_in[14];
    const float* bfh21  = (const float*)d_in[15];
    const float* Wfh22  = (const float*)d_in[16];
    const float* bfh22  = (const float*)d_in[17];
    const float* Wl1    = (const float*)d_in[18];
    const float* bl1    = (const float*)d_in[19];
    const float* Wl2    = (const float*)d_in[20];
    const float* bl2    = (const float*)d_in[21];
    float* out = (float*)d_out;

    // ---- workspace carve-out (~272 MB) -----------------------------------
    char* ws = (char*)d_ws;
    size_t off = 0;
    auto carve = [&](size_t bytes) -> float* {
        float* p = (float*)(ws + off);
        off = (off + bytes + 255) & ~(size_t)255;
        return p;
    };
    float* Wt    = carve((size_t)KROWS * NCOLS * sizeof(float));       // 3.1 MB
    float* biasc = carve((size_t)NCOLS * sizeof(float));
    float* zerox = carve((size_t)DINP * sizeof(float));
    float* h0    = carve((size_t)BATCH * LEAVES * HID * sizeof(float)); // 64 MB
    float* c0    = carve((size_t)BATCH * LEAVES * HID * sizeof(float));
    float* h1    = carve((size_t)BATCH * LEAVES * HID * sizeof(float));
    float* c1    = carve((size_t)BATCH * LEAVES * HID * sizeof(float));
    (void)ws_size; (void)in_sizes; (void)n_in; (void)out_size;

    // ---- pack weights ----------------------------------------------------
    {
        int total = KROWS * NCOLS;
        pack_kernel<<<(total + 255) / 256, 256, 0, stream>>>(
            Wioux, bioux, Wiouh1, biouh1, Wiouh2, biouh2,
            Wfx, bfx, Wfh11, bfh11, Wfh12, bfh12,
            Wfh21, bfh21, Wfh22, bfh22, Wt, biasc, zerox);
    }

    // ---- leaf level: x-only, 768 cols (iou), h=c=0 -> c = i*u ------------
    {
        const int M = BATCH * LEAVES;   // 65536 rows -> 4096 blocks
        fused_level_kernel<6, false><<<M / 16, 256, 0, stream>>>(
            input_ids, emb, zerox, h0 /*unused*/, c0 /*unused*/,
            Wt, biasc, h0, c0,
            /*node_off=*/0, /*log2n=*/8, /*n_nodes=*/LEAVES);
    }

    // ---- internal levels (serial dependency chain) -----------------------
    const int offs[8] = {256, 384, 448, 480, 496, 504, 508, 510};
    const int ns[8]   = {128,  64,  32,  16,   8,   4,   2,   1};
    float* hp = h0; float* cp = c0; float* hn = h1; float* cn = c1;
    for (int lv = 0; lv < 8; ++lv) {
        const int n  = ns[lv];
        const int M  = BATCH * n;
        const int lg = 7 - lv;
        fused_level_kernel<10, true><<<M / 16, 256, 0, stream>>>(
            input_ids, emb, zerox, hp, cp, Wt, biasc, hn, cn,
            offs[lv], lg, n);
        float* th = hp; hp = hn; hn = th;
        float* tc = cp; cp = cn; cn = tc;
    }

    // ---- head ------------------------------------------------------------
    head_kernel<<<BATCH, 128, 0, stream>>>(hp, Wl1, bl1, Wl2, bl2, out);
}